// FlowMatchingLoss_69672959475775
// MI455X (gfx1250) — compile-verified
//
#include <hip/hip_runtime.h>
#include <math.h>

typedef __attribute__((ext_vector_type(2))) float v2f;
typedef __attribute__((ext_vector_type(8))) float v8f;

#define BB 8
#define MM 2048
#define KK 2048
#define INV_REG_OT 20.0f          // 1/0.05
#define TAUF 0.9523809523809523f  // 1.0/(1.0+0.05)
#define NEG_BIG -3.0e38f

// ---- workspace layout (float indices) ----
#define WS_A      0        // a            [B*M]
#define WS_LOGA   16384    // log_a        [B*M]
#define WS_LOGU   32768    // log_u        [B*M]
#define WS_SURV   49152    // surv         [B*M]
#define WS_S2     65536    // |src|^2      [B*M]
#define WS_T2     81920    // |tgt|^2      [B*K]
#define WS_LOGV   98304    // log_v        [B*K]
#define WS_LOGB   114688   // log_b        [B]
#define WS_CMAX   114696   // cost max bits[B] (uint)
#define WS_INVM   114704   // 1/(max+eps)  [B]
#define WS_SUMA   114712   // sum(a)       [B]
#define WS_SCAL   114720   // vel_num, vel_den, bce_sum, sum_p, sum_pc
#define WS_VMAX   114728   // per-batch shift for LSE [B]

// ds_swizzle group-of-32 xor-lane exchange: offset = xor<<10 | and_mask(0x1f)
#define SWZ(v, imm) __int_as_float(__builtin_amdgcn_ds_swizzle(__float_as_int(v), (imm)))

__device__ inline float half16_sum(float v) {   // sum across lanes 0-15 / 16-31
    v += SWZ(v, 0x041f);   // xor 1
    v += SWZ(v, 0x081f);   // xor 2
    v += SWZ(v, 0x101f);   // xor 4
    v += SWZ(v, 0x201f);   // xor 8
    return v;
}
__device__ inline float wave_sum(float v) {
    v = half16_sum(v);
    v += SWZ(v, 0x401f);   // xor 16
    return v;
}
__device__ inline float wave_max(float v) {
    v = fmaxf(v, SWZ(v, 0x041f));
    v = fmaxf(v, SWZ(v, 0x081f));
    v = fmaxf(v, SWZ(v, 0x101f));
    v = fmaxf(v, SWZ(v, 0x201f));
    v = fmaxf(v, SWZ(v, 0x401f));
    return v;
}

// Unconditional 3-component load, then lane-half select (no divergent loads).
// lanes 0-15: (x,y) = K0,K1 ; lanes 16-31: (z,0) = K2,K3
__device__ inline v2f load_frag3(const float* __restrict__ pts, int hi) {
    float x = pts[0];
    float y = pts[1];
    float z = pts[2];
    v2f f;
    f.x = hi ? z : x;
    f.y = hi ? 0.0f : y;
    return f;
}

// ---------------- zero scalars ----------------
__global__ void fm_zero_kernel(float* ws) {
    int i = threadIdx.x;
    if (i < 32) ws[WS_CMAX + i] = 0.0f;   // cmax, invm, sumA, scalars
}

// ---------------- per-point init ----------------
__global__ void fm_init_kernel(const float* __restrict__ x0,
                               const float* __restrict__ xgt,
                               const float* __restrict__ vpred,
                               const float* __restrict__ alpha,
                               const float* __restrict__ matched,
                               float* ws) {
    int i = blockIdx.x * blockDim.x + threadIdx.x;   // grid sized exactly B*M
    int b = i / MM;
    int lane = threadIdx.x & 31;

    float al = alpha[i];
    float asoft = 10.0f * tanhf(al * 0.1f);
    float a = 1.0f / (1.0f + __expf(-asoft));
    ws[WS_A + i]    = a;
    ws[WS_LOGA + i] = __logf(fmaxf(a, 1e-30f));

    const float* p = x0 + (size_t)i * 3;
    ws[WS_S2 + i] = p[0]*p[0] + p[1]*p[1] + p[2]*p[2];
    const float* q = xgt + (size_t)i * 3;   // B*K == B*M
    ws[WS_T2 + i] = q[0]*q[0] + q[1]*q[1] + q[2]*q[2];

    // velocity head
    const float* vp = vpred + (size_t)i * 3;
    const float* mt = matched + (size_t)i * 3;
    float d0 = vp[0] - (mt[0] - p[0]);
    float d1 = vp[1] - (mt[1] - p[1]);
    float d2 = vp[2] - (mt[2] - p[2]);
    float se = d0*d0 + d1*d1 + d2*d2;
    float w  = 1.0f / (1.0f + __expf(-al));

    float ra   = wave_sum(a);
    float rnum = wave_sum(se * w);
    float rden = wave_sum(w);
    if (lane == 0) {
        atomicAdd(&ws[WS_SUMA + b], ra);
        atomicAdd(&ws[WS_SCAL + 0], rnum);
        atomicAdd(&ws[WS_SCAL + 1], rden);
    }

    ws[WS_LOGU + i] = 0.0f;
    ws[WS_LOGV + i] = 0.0f;
}

// ---------------- cost max (WMMA) ----------------
__global__ void fm_costmax_kernel(const float* __restrict__ P,
                                  const float* __restrict__ Q,
                                  const float* __restrict__ s2P,
                                  const float* __restrict__ t2Q,
                                  float* ws) {
    const int tiles_m = MM / 16;
    int b      = blockIdx.x / tiles_m;
    int base_m = (blockIdx.x % tiles_m) * 16;
    int wave = threadIdx.x >> 5, lane = threadIdx.x & 31;
    int hi = lane >> 4, l16 = lane & 15;

    const float* Pb  = P   + (size_t)b * MM * 3;
    const float* Qb  = Q   + (size_t)b * KK * 3;
    const float* s2b = s2P + (size_t)b * MM;
    const float* t2b = t2Q + (size_t)b * KK;

    v2f A = load_frag3(Pb + (base_m + l16) * 3, hi);
    int m0 = base_m + 8 * hi;
    float s2r[8];
#pragma unroll
    for (int r = 0; r < 8; ++r) s2r[r] = s2b[m0 + r];

    float cm = 0.0f;
    for (int ct = wave; ct < KK / 16; ct += 8) {   // 2 tiles per iteration
        int n0 = ct * 16;
        int n1 = n0 + 64;
        v2f B0 = load_frag3(Qb + (n0 + l16) * 3, hi);
        v2f B1 = load_frag3(Qb + (n1 + l16) * 3, hi);
        float t20 = t2b[n0 + l16];
        float t21 = t2b[n1 + l16];
        v8f a0 = {}, a1 = {};
        a0 = __builtin_amdgcn_wmma_f32_16x16x4_f32(false, A, false, B0, (short)0, a0, false, false);
        a1 = __builtin_amdgcn_wmma_f32_16x16x4_f32(false, A, false, B1, (short)0, a1, false, false);
#pragma unroll
        for (int r = 0; r < 8; ++r) {
            cm = fmaxf(cm, fmaxf(s2r[r] + t20 - 2.0f * a0[r], 0.0f));
            cm = fmaxf(cm, fmaxf(s2r[r] + t21 - 2.0f * a1[r], 0.0f));
        }
    }
    cm = wave_max(cm);
    __shared__ float wmx[4];
    if (lane == 0) wmx[wave] = cm;
    __syncthreads();
    if (threadIdx.x == 0) {
        float m = fmaxf(fmaxf(wmx[0], wmx[1]), fmaxf(wmx[2], wmx[3]));
        atomicMax((unsigned int*)&ws[WS_CMAX + b], __float_as_uint(m));
    }
}

// ---------------- per-batch scalars ----------------
__global__ void fm_batch_kernel(float* ws) {
    int b = threadIdx.x;
    if (b >= BB) return;
    ws[WS_LOGB + b] = __logf(fmaxf(ws[WS_SUMA + b] * (1.0f / (float)KK), 1e-30f));
    float mx = __uint_as_float(*(unsigned int*)&ws[WS_CMAX + b]);
    ws[WS_INVM + b] = 1.0f / (mx + 1e-8f);
}

// ---------------- per-batch max of vec (LSE shift) ----------------
__global__ void fm_vecmax_kernel(const float* __restrict__ vec, int C, float* ws) {
    int b = blockIdx.x;
    int tid = threadIdx.x, lane = tid & 31, wave = tid >> 5;
    float m = NEG_BIG;
    for (int i = tid; i < C; i += blockDim.x)
        m = fmaxf(m, vec[(size_t)b * C + i]);
    m = wave_max(m);
    __shared__ float wm[8];
    if (lane == 0) wm[wave] = m;
    __syncthreads();
    if (tid == 0) {
        float r = wm[0];
        for (int w = 1; w < (int)(blockDim.x >> 5); ++w) r = fmaxf(r, wm[w]);
        ws[WS_VMAX + b] = r;
    }
}

// ---------------- Sinkhorn half step (WMMA + shifted LSE) ----------------
// out[r] = TAU * (marg[r] - logsumexp_c( -cost_norm(r,c)/reg + vec[c] ))
// shift = max_c vec[c] precomputed -> single-shift LSE is numerically safe.
__global__ void fm_half_kernel(const float* __restrict__ P,
                               const float* __restrict__ Q,
                               const float* __restrict__ s2P,
                               const float* __restrict__ t2Q,
                               const float* __restrict__ vec,
                               const float* __restrict__ marg, int margScalar,
                               const float* __restrict__ invm,
                               const float* __restrict__ vshift,
                               float* __restrict__ outv,
                               int R, int C) {
    const int tiles_m = R / 16;
    int b      = blockIdx.x / tiles_m;
    int base_m = (blockIdx.x % tiles_m) * 16;
    int wave = threadIdx.x >> 5, lane = threadIdx.x & 31;
    int hi = lane >> 4, l16 = lane & 15;

    const float* Pb   = P   + (size_t)b * R * 3;
    const float* Qb   = Q   + (size_t)b * C * 3;
    const float* s2b  = s2P + (size_t)b * R;
    const float* t2b  = t2Q + (size_t)b * C;
    const float* vecb = vec + (size_t)b * C;

    v2f A = load_frag3(Pb + (base_m + l16) * 3, hi);
    int m0 = base_m + 8 * hi;
    float s2r[8];
#pragma unroll
    for (int r = 0; r < 8; ++r) s2r[r] = s2b[m0 + r];
    float inv   = invm[b];
    float shift = vshift[b];

    float sm[8];
#pragma unroll
    for (int r = 0; r < 8; ++r) sm[r] = 0.0f;

    for (int ct = wave; ct < C / 16; ct += 8) {   // 2 tiles per iteration
        int n0 = ct * 16;
        int n1 = n0 + 64;
        v2f B0 = load_frag3(Qb + (n0 + l16) * 3, hi);
        v2f B1 = load_frag3(Qb + (n1 + l16) * 3, hi);
        float t20 = t2b[n0 + l16];
        float t21 = t2b[n1 + l16];
        float v0  = vecb[n0 + l16] - shift;
        float v1  = vecb[n1 + l16] - shift;
        v8f a0 = {}, a1 = {};
        a0 = __builtin_amdgcn_wmma_f32_16x16x4_f32(false, A, false, B0, (short)0, a0, false, false);
        a1 = __builtin_amdgcn_wmma_f32_16x16x4_f32(false, A, false, B1, (short)0, a1, false, false);
#pragma unroll
        for (int r = 0; r < 8; ++r) {
            float c0 = fmaxf(s2r[r] + t20 - 2.0f * a0[r], 0.0f) * inv;
            float c1 = fmaxf(s2r[r] + t21 - 2.0f * a1[r], 0.0f) * inv;
            sm[r] += __expf(v0 - c0 * INV_REG_OT);
            sm[r] += __expf(v1 - c1 * INV_REG_OT);
        }
    }
#pragma unroll
    for (int r = 0; r < 8; ++r) sm[r] = half16_sum(sm[r]);

    __shared__ float lsm[4][16];
    if (l16 == 0) {
#pragma unroll
        for (int r = 0; r < 8; ++r) lsm[wave][8 * hi + r] = sm[r];
    }
    __syncthreads();
    if (threadIdx.x < 16) {
        int row = threadIdx.x;
        float S = lsm[0][row] + lsm[1][row] + lsm[2][row] + lsm[3][row];
        float lse = shift + __logf(S);
        float mg = margScalar ? marg[b] : marg[(size_t)b * R + base_m + row];
        outv[(size_t)b * R + base_m + row] = TAUF * (mg - lse);
    }
}

// ---------------- final pi pass (WMMA): surv, sum(pi), sum(pi*cost) ----------------
__global__ void fm_pi_kernel(const float* __restrict__ P,
                             const float* __restrict__ Q,
                             const float* __restrict__ s2P,
                             const float* __restrict__ t2Q,
                             const float* __restrict__ lu,
                             const float* __restrict__ lv,
                             const float* __restrict__ invm,
                             float* ws) {
    const int tiles_m = MM / 16;
    int b      = blockIdx.x / tiles_m;
    int base_m = (blockIdx.x % tiles_m) * 16;
    int wave = threadIdx.x >> 5, lane = threadIdx.x & 31;
    int hi = lane >> 4, l16 = lane & 15;

    const float* Pb  = P   + (size_t)b * MM * 3;
    const float* Qb  = Q   + (size_t)b * KK * 3;
    const float* s2b = s2P + (size_t)b * MM;
    const float* t2b = t2Q + (size_t)b * KK;
    const float* lvb = lv  + (size_t)b * KK;

    v2f A = load_frag3(Pb + (base_m + l16) * 3, hi);
    int m0 = base_m + 8 * hi;
    float s2r[8], lur[8];
#pragma unroll
    for (int r = 0; r < 8; ++r) {
        s2r[r] = s2b[m0 + r];
        lur[r] = lu[(size_t)b * MM + m0 + r];
    }
    float inv = invm[b];

    float srow[8];
#pragma unroll
    for (int r = 0; r < 8; ++r) srow[r] = 0.0f;
    float sp = 0.0f, spc = 0.0f;

    for (int ct = wave; ct < KK / 16; ct += 8) {   // 2 tiles per iteration
        int n0 = ct * 16;
        int n1 = n0 + 64;
        v2f B0 = load_frag3(Qb + (n0 + l16) * 3, hi);
        v2f B1 = load_frag3(Qb + (n1 + l16) * 3, hi);
        float t20 = t2b[n0 + l16];
        float t21 = t2b[n1 + l16];
        float w0  = lvb[n0 + l16];
        float w1  = lvb[n1 + l16];
        v8f a0 = {}, a1 = {};
        a0 = __builtin_amdgcn_wmma_f32_16x16x4_f32(false, A, false, B0, (short)0, a0, false, false);
        a1 = __builtin_amdgcn_wmma_f32_16x16x4_f32(false, A, false, B1, (short)0, a1, false, false);
#pragma unroll
        for (int r = 0; r < 8; ++r) {
            float c0 = fmaxf(s2r[r] + t20 - 2.0f * a0[r], 0.0f) * inv;
            float c1 = fmaxf(s2r[r] + t21 - 2.0f * a1[r], 0.0f) * inv;
            float p0 = __expf(lur[r] - c0 * INV_REG_OT + w0);
            float p1 = __expf(lur[r] - c1 * INV_REG_OT + w1);
            srow[r] += p0 + p1;
            sp  += p0 + p1;
            spc += p0 * c0 + p1 * c1;
        }
    }
#pragma unroll
    for (int r = 0; r < 8; ++r) srow[r] = half16_sum(srow[r]);
    sp  = wave_sum(sp);
    spc = wave_sum(spc);

    __shared__ float lrow[4][16];
    __shared__ float lsp[4], lspc[4];
    if (l16 == 0) {
#pragma unroll
        for (int r = 0; r < 8; ++r) lrow[wave][8 * hi + r] = srow[r];
    }
    if (lane == 0) { lsp[wave] = sp; lspc[wave] = spc; }
    __syncthreads();
    if (threadIdx.x < 16) {
        int row = threadIdx.x;
        float s = lrow[0][row] + lrow[1][row] + lrow[2][row] + lrow[3][row];
        ws[WS_SURV + (size_t)b * MM + base_m + row] = s;
    }
    if (threadIdx.x == 0) {
        atomicAdd(&ws[WS_SCAL + 3], lsp[0] + lsp[1] + lsp[2] + lsp[3]);
        atomicAdd(&ws[WS_SCAL + 4], lspc[0] + lspc[1] + lspc[2] + lspc[3]);
    }
}

// ---------------- BCE ----------------
__global__ void fm_bce_kernel(const float* __restrict__ alpha, float* ws) {
    int i = blockIdx.x * blockDim.x + threadIdx.x;   // grid sized exactly B*M
    int lane = threadIdx.x & 31;
    float x = alpha[i];
    float a = ws[WS_A + i];
    float y = fminf(fmaxf(ws[WS_SURV + i] / (a + 1e-8f), 0.0f), 1.0f);
    float bce = fmaxf(x, 0.0f) - x * y + log1pf(__expf(-fabsf(x)));
    float r = wave_sum(bce);
    if (lane == 0) atomicAdd(&ws[WS_SCAL + 2], r);
}

// ---------------- combine ----------------
__global__ void fm_combine_kernel(const float* ws, float* out) {
    float vel = ws[WS_SCAL + 0] / fmaxf(ws[WS_SCAL + 1], 1.0f);
    float bce = ws[WS_SCAL + 2] * (1.0f / (float)(BB * MM));
    float ot  = ws[WS_SCAL + 4] / fmaxf(ws[WS_SCAL + 3], 1e-8f);
    out[0] = 1.0f * vel + 1.0f * bce + 0.1f * ot;
}

extern "C" void kernel_launch(void* const* d_in, const int* in_sizes, int n_in,
                              void* d_out, int out_size, void* d_ws, size_t ws_size,
                              hipStream_t stream) {
    const float* x0      = (const float*)d_in[0];  // (B,M,3)
    const float* xgt     = (const float*)d_in[1];  // (B,K,3)
    const float* vpred   = (const float*)d_in[2];  // (B,M,3)
    const float* alpha   = (const float*)d_in[3];  // (B,M,1)
    const float* matched = (const float*)d_in[5];  // (B,M,3)
    float* ws  = (float*)d_ws;
    float* out = (float*)d_out;

    const int nPts     = BB * MM;
    const int tileGrid = BB * (MM / 16);   // 1024 blocks

    fm_zero_kernel<<<1, 64, 0, stream>>>(ws);
    fm_init_kernel<<<nPts / 256, 256, 0, stream>>>(x0, xgt, vpred, alpha, matched, ws);
    fm_costmax_kernel<<<tileGrid, 128, 0, stream>>>(x0, xgt, ws + WS_S2, ws + WS_T2, ws);
    fm_batch_kernel<<<1, 64, 0, stream>>>(ws);

    for (int it = 0; it < 5; ++it) {
        // log_u = tau*(log_a - lse_k(K_log + log_v))
        fm_vecmax_kernel<<<BB, 256, 0, stream>>>(ws + WS_LOGV, KK, ws);
        fm_half_kernel<<<tileGrid, 128, 0, stream>>>(
            x0, xgt, ws + WS_S2, ws + WS_T2,
            ws + WS_LOGV, ws + WS_LOGA, 0, ws + WS_INVM, ws + WS_VMAX,
            ws + WS_LOGU, MM, KK);
        // log_v = tau*(log_b - lse_m(K_log + log_u))   (transposed roles)
        fm_vecmax_kernel<<<BB, 256, 0, stream>>>(ws + WS_LOGU, MM, ws);
        fm_half_kernel<<<tileGrid, 128, 0, stream>>>(
            xgt, x0, ws + WS_T2, ws + WS_S2,
            ws + WS_LOGU, ws + WS_LOGB, 1, ws + WS_INVM, ws + WS_VMAX,
            ws + WS_LOGV, KK, MM);
    }

    fm_pi_kernel<<<tileGrid, 128, 0, stream>>>(
        x0, xgt, ws + WS_S2, ws + WS_T2, ws + WS_LOGU, ws + WS_LOGV, ws + WS_INVM, ws);
    fm_bce_kernel<<<nPts / 256, 256, 0, stream>>>(alpha, ws);
    fm_combine_kernel<<<1, 1, 0, stream>>>(ws, out);
}